// AnalyticBlock_47974784697009
// MI455X (gfx1250) — compile-verified
//
#include <hip/hip_runtime.h>
#include <math.h>

#define Cc   256
#define Hh   128
#define Ww   128
#define Bb   16

typedef __attribute__((ext_vector_type(2))) float v2f;
typedef __attribute__((ext_vector_type(8))) float v8f;

// ws layout (floats):
//   [0..255]     wlo  (W[c][c-1])
//   [256..511]   wd   (W[c][c] + 1e-3)
//   [512..767]   whi  (W[c][c+1])
//   [768] a_gm  [769] a_lap  [770] gamma  [771] 1-gamma
//   [1024..1279] gsum   [1280..1535] gssq
//   [1536..1791] A_c    [1792..2047] B_c

__device__ __forceinline__ float softplus_f(float x) {
  return (x > 20.f) ? x : log1pf(expf(x));
}

__global__ void k_prep(const float* __restrict__ P,
                       const float* __restrict__ alpha_gm,
                       const float* __restrict__ alpha_lap,
                       const float* __restrict__ gamma_p,
                       float* __restrict__ ws) {
  int c = threadIdx.x;                       // 256 threads
  float wlo = (c > 0)      ? softplus_f(P[c * Cc + (c - 1)]) : 0.f;
  float whi = (c < Cc - 1) ? softplus_f(P[c * Cc + (c + 1)]) : 0.f;
  float wd  = softplus_f(P[c * Cc + c]) + 1e-3f;
  ws[c]          = wlo;
  ws[256 + c]    = wd;
  ws[512 + c]    = whi;
  ws[1024 + c]   = 0.f;    // gsum  (re-zero every call: deterministic replays)
  ws[1280 + c]   = 0.f;    // gssq
  if (c == 0) {
    ws[768] = softplus_f(alpha_gm[0]);
    ws[769] = softplus_f(alpha_lap[0]);
    float g = 1.f / (1.f + expf(-gamma_p[0]));
    ws[770] = g;
    ws[771] = 1.f - g;
  }
}

__global__ void k_finalize(const float* __restrict__ bn_w,
                           const float* __restrict__ bn_b,
                           float* __restrict__ ws) {
  int c = threadIdx.x;                       // 256 threads
  const float N = (float)Bb * (float)Hh * (float)Ww;
  float mean = ws[1024 + c] / N;
  float var  = ws[1280 + c] / N - mean * mean;
  float rstd = rsqrtf(var + 1e-5f);
  float g     = ws[770];
  float scale = rstd * bn_w[c];
  ws[1536 + c] = g * scale;                        // A_c
  ws[1792 + c] = g * (bn_b[c] - mean * scale);     // B_c
}

// Fused stencil + WMMA banded mix. PASS==1: accumulate stats. PASS==2: write out.
// Workgroup: 256 threads (8 waves). Spatial tile: 32 (w) x 16 (h), halo-staged in LDS.
template <int PASS>
__global__ void k_main(const float* __restrict__ x,
                       float* __restrict__ out,
                       float* __restrict__ ws) {
  __shared__ float xs[18 * 612];    // 18 channels x (18 rows * 34 cols), 44 KB
  __shared__ float red[256];        // [0..127] sums, [128..255] sumsq : [wave][ch]

  const int tid  = threadIdx.x;
  const int lane = tid & 31;
  const int wv   = tid >> 5;            // wave 0..7
  const int p    = lane & 15;           // pixel index within the wave's 16-pixel group
  const bool hi  = (lane >= 16);

  const int b  = blockIdx.z;
  const int h0 = blockIdx.y * 16;
  const int w0 = blockIdx.x * 32;

  const int col     = ((wv & 1) << 4) + p;   // 0..31 in tile
  const int rowbase = (wv >> 1);             // row = g*4 + rowbase

  const float a_gm    = ws[768];
  const float a_lap   = ws[769];
  const float one_m_g = ws[771];

  for (int k = 0; k < 16; ++k) {
    const int c0 = k * 16;
    __syncthreads();
    // ---- stage x[c0-1 .. c0+16] tile (34x18 halo) into LDS ----
    for (int idx = tid; idx < 18 * 612; idx += 256) {
      int chl = idx / 612;
      int rem = idx - chl * 612;
      int r   = rem / 34;
      int cc  = rem - r * 34;
      int gch = c0 - 1 + chl;
      int gh  = h0 + r - 1;
      int gw  = w0 + cc - 1;
      float v = 0.f;
      if ((unsigned)gch < (unsigned)Cc && (unsigned)gh < (unsigned)Hh &&
          (unsigned)gw < (unsigned)Ww)
        v = x[(((size_t)b * Cc + gch) * Hh + gh) * Ww + gw];
      xs[idx] = v;
    }
    __syncthreads();

    // ---- A matrix (16x16 block-diagonal of tridiagonal W), per ISA f32 A layout ----
    const int m = p;                               // output row within block
    const float wlo_m = ws[c0 + m];
    const float wd_m  = ws[256 + c0 + m];
    const float whi_m = ws[512 + c0 + m];
    const int kbase = hi ? 2 : 0;                  // lanes 0-15: K{0,1}; 16-31: K{2,3}
    auto wval = [&](int kcol) -> float {
      return (kcol == m) ? wd_m : (kcol == m - 1) ? wlo_m
                                : (kcol == m + 1) ? whi_m : 0.f;
    };
    v2f a0 = { wval(kbase + 0),  wval(kbase + 1)  };
    v2f a1 = { wval(kbase + 4),  wval(kbase + 5)  };
    v2f a2 = { wval(kbase + 8),  wval(kbase + 9)  };
    v2f a3 = { wval(kbase + 12), wval(kbase + 13) };
    const float wlo_c0  = hi ? 0.f : ws[c0];            // corner W[c0][c0-1]
    const float whi_c15 = hi ? ws[512 + c0 + 15] : 0.f; // corner W[c0+15][c0+16]

    float s[8], q[8];
#pragma unroll
    for (int i = 0; i < 8; ++i) { s[i] = 0.f; q[i] = 0.f; }

    for (int g = 0; g < 4; ++g) {
      const int r = g * 4 + rowbase;               // interior row 0..15
      auto featc = [&](int chl) -> float {
        const float* q0 = &xs[chl * 612 + r * 34 + col];
        float x00 = q0[0],  x01 = q0[1],  x02 = q0[2];
        float x10 = q0[34], x11 = q0[35], x12 = q0[36];
        float x20 = q0[68], x21 = q0[69], x22 = q0[70];
        float gx  = (x00 - x02) + 2.f * (x10 - x12) + (x20 - x22);
        float gy  = (x00 + 2.f * x01 + x02) - (x20 + 2.f * x21 + x22);
        float lap = fabsf(x01 + x10 + x12 + x21 - 4.f * x11);
        // operand >= 1e-6: never denormal -> raw v_sqrt_f32 (1 ULP), skip the
        // IEEE denorm pre-scale/fixup sequence sqrtf() emits.
        float gm  = __builtin_amdgcn_sqrtf(gx * gx + gy * gy + 1e-6f);
        return fmaxf(a_gm * gm + a_lap * lap, 0.f);
      };
      // B matrix: feat for 16 channels x 16 pixels (f32 K=4 B layout)
      const int cb = 1 + kbase;                    // LDS channel of this lane's K=0
      v2f b0 = { featc(cb + 0),  featc(cb + 1)  };
      v2f b1 = { featc(cb + 4),  featc(cb + 5)  };
      v2f b2 = { featc(cb + 8),  featc(cb + 9)  };
      v2f b3 = { featc(cb + 12), featc(cb + 13) };
      const float fb = featc(hi ? 17 : 0);         // boundary feat (c0-1 or c0+16)

      v8f d = {};
      d = __builtin_amdgcn_wmma_f32_16x16x4_f32(false, a0, false, b0, (short)0, d, false, false);
      d = __builtin_amdgcn_wmma_f32_16x16x4_f32(false, a1, false, b1, (short)0, d, false, false);
      d = __builtin_amdgcn_wmma_f32_16x16x4_f32(false, a2, false, b2, (short)0, d, false, false);
      d = __builtin_amdgcn_wmma_f32_16x16x4_f32(false, a3, false, b3, (short)0, d, false, false);
      d[0] += wlo_c0  * fb;   // mixed[c0]    += W[c0][c0-1]    * feat[c0-1]
      d[7] += whi_c15 * fb;   // mixed[c0+15] += W[c0+15][c0+16]* feat[c0+16]

      if (PASS == 1) {
#pragma unroll
        for (int i = 0; i < 8; ++i) { s[i] += d[i]; q[i] += d[i] * d[i]; }
      } else {
        const int h = h0 + r;
        const int wcol = w0 + col;
#pragma unroll
        for (int i = 0; i < 8; ++i) {
          int M  = i + (hi ? 8 : 0);
          int ch = c0 + M;
          float Ac = ws[1536 + ch], Bc = ws[1792 + ch];
          float xv = xs[(1 + M) * 612 + (r + 1) * 34 + (col + 1)];
          out[(((size_t)b * Cc + ch) * Hh + h) * Ww + wcol] =
              one_m_g * xv + Ac * d[i] + Bc;
        }
      }
    }

    if (PASS == 1) {
      // reduce over the 16 pixels within each half-wave
#pragma unroll
      for (int i = 0; i < 8; ++i) {
        s[i] += __shfl_xor(s[i], 1); s[i] += __shfl_xor(s[i], 2);
        s[i] += __shfl_xor(s[i], 4); s[i] += __shfl_xor(s[i], 8);
        q[i] += __shfl_xor(q[i], 1); q[i] += __shfl_xor(q[i], 2);
        q[i] += __shfl_xor(q[i], 4); q[i] += __shfl_xor(q[i], 8);
      }
      if (lane == 0 || lane == 16) {
        int off = hi ? 8 : 0;
#pragma unroll
        for (int i = 0; i < 8; ++i) {
          red[wv * 16 + off + i]       = s[i];
          red[128 + wv * 16 + off + i] = q[i];
        }
      }
      __syncthreads();
      if (tid < 32) {           // cross-wave reduce, one atomic per channel
        int chn = tid & 15;
        int sel = (tid < 16) ? 0 : 128;
        float acc = 0.f;
#pragma unroll
        for (int w2 = 0; w2 < 8; ++w2) acc += red[sel + w2 * 16 + chn];
        atomicAdd(&ws[((tid < 16) ? 1024 : 1280) + c0 + chn], acc);
      }
    }
  }
}

extern "C" void kernel_launch(void* const* d_in, const int* in_sizes, int n_in,
                              void* d_out, int out_size, void* d_ws, size_t ws_size,
                              hipStream_t stream) {
  const float* x         = (const float*)d_in[0];
  const float* P         = (const float*)d_in[1];
  const float* alpha_gm  = (const float*)d_in[2];
  const float* alpha_lap = (const float*)d_in[3];
  const float* gamma_p   = (const float*)d_in[4];
  const float* bn_w      = (const float*)d_in[5];
  const float* bn_b      = (const float*)d_in[6];
  float* out = (float*)d_out;
  float* ws  = (float*)d_ws;

  k_prep<<<1, 256, 0, stream>>>(P, alpha_gm, alpha_lap, gamma_p, ws);
  dim3 grid(Ww / 32, Hh / 16, Bb);
  k_main<1><<<grid, 256, 0, stream>>>(x, out, ws);     // stats pass
  k_finalize<<<1, 256, 0, stream>>>(bn_w, bn_b, ws);
  k_main<2><<<grid, 256, 0, stream>>>(x, out, ws);     // output pass
}